// WaveCell_14894946582958
// MI455X (gfx1250) — compile-verified
//
#include <hip/hip_runtime.h>
#include <stdint.h>

// 2-D FDTD wave recurrence, 300 sequential steps (grid sync = kernel boundary).
// CDNA5 paths: TDM tensor_load_to_lds (stage un1 tile, TENSORcnt),
// global_store_async_from_lds_b128 th:TH_STORE_NT (315 MB output stream,
// ASYNCcnt), s_wait_tensorcnt / s_wait_asynccnt, global_prefetch_b8.

#define NXc   256
#define NYc   256
#define TILE  16
#define BATCH 4

typedef uint32_t v4u __attribute__((ext_vector_type(4)));
typedef uint32_t v8u __attribute__((ext_vector_type(8)));

// One 2-D TDM DMA: nrows x 256 f32 tile, row stride 256 f32, global -> LDS.
__device__ __forceinline__ void tdm_load_tile(uint32_t lds_byte_addr,
                                              uint64_t gaddr, uint32_t nrows) {
  v4u g0;
  g0.x = 1u;                                   // count=1 valid, no gather
  g0.y = lds_byte_addr;                        // lds_addr [63:32]
  g0.z = (uint32_t)gaddr;                      // global_addr [95:64]
  g0.w = ((uint32_t)(gaddr >> 32) & 0x01FFFFFFu) | 0x80000000u; // addr hi | type=2
  v8u g1;
  g1.s0 = 2u << 16;                            // data_size=2 (4B); mask=0; no pad
  g1.s1 = (uint32_t)NYc << 16;                 // tensor_dim0 = 256 (low 16)
  g1.s2 = nrows << 16;                         // tensor_dim1 = nrows (low 16)
  g1.s3 = (uint32_t)NYc << 16;                 // tile_dim0 = 256
  g1.s4 = nrows;                               // tile_dim1 = nrows (2D tile)
  g1.s5 = (uint32_t)NYc;                       // tensor_dim0_stride = 256
  g1.s6 = 0u;
  g1.s7 = 0u;
  asm volatile("tensor_load_to_lds %0, %1" :: "s"(g0), "s"(g1) : "memory");
}

__device__ __forceinline__ void async_store_nt_b128(uint64_t gaddr, uint32_t lds_off) {
  asm volatile("global_store_async_from_lds_b128 %0, %1, off th:TH_STORE_NT"
               :: "v"(gaddr), "v"(lds_off) : "memory");
}
__device__ __forceinline__ void wait_asynccnt0() {
  asm volatile("s_wait_asynccnt 0" ::: "memory");
}

__global__ __launch_bounds__(NYc)
void WaveCell_step_kernel(const float* un1,        // state t-1 (read)
                          const float* un2,        // state t-2 (read)
                          float*       unw,        // state write (= un2 buffer)
                          float* __restrict__ out, // (B,T,NX,NY)
                          const float* __restrict__ A1,
                          const float* __restrict__ A3,
                          const float* __restrict__ c2,
                          const float* __restrict__ xsrc,   // (B,T)
                          const float* __restrict__ A2p,    // scalar
                          const float* __restrict__ invh2p, // scalar
                          const int* __restrict__ sxp,
                          const int* __restrict__ syp,
                          int t, int T)
{
  __shared__ __align__(16) float sh[(TILE + 2) * NYc]; // un1 tile + halo rows
  __shared__ __align__(16) float sres[TILE * NYc];     // staged output tile

  const int c  = threadIdx.x;   // column 0..255 (one lane per column)
  const int rb = blockIdx.x;    // row block 0..15
  const int b  = blockIdx.y;    // batch
  const int x0 = rb * TILE;
  const size_t plane = (size_t)NXc * NYc;

  const float* u1b  = un1 + (size_t)b * plane;
  const float* un2b = un2 + (size_t)b * plane;
  float*       unwb = unw + (size_t)b * plane;

  // Warm caches for the direct global reads below.
  __builtin_prefetch(un2b + (size_t)x0 * NYc + c, 0, 3);
  __builtin_prefetch(A1   + (size_t)x0 * NYc + c, 0, 3);

  // Row range clamped to the grid; halo rows outside get explicit zeros.
  const int gx0    = x0 - 1;
  const int gstart = (gx0 > 0) ? gx0 : 0;
  const int gend   = (x0 + TILE < NXc - 1) ? (x0 + TILE) : (NXc - 1); // inclusive
  const int nrows  = gend - gstart + 1;   // 17 or 18
  const int lrow0  = gstart - gx0;        // 1 iff top halo missing

  const uint32_t shbase = (uint32_t)(size_t)(&sh[0]); // low 32 bits = LDS offset
  if (lrow0)                sh[c]                      = 0.0f; // top halo zero
  if (gend < x0 + TILE)     sh[(TILE + 1) * NYc + c]   = 0.0f; // bottom halo zero

  // One TDM descriptor per workgroup; wave 0 issues it (TDM ignores EXEC, so
  // keep other waves branched around it), waits TENSORcnt, then barrier.
  if (threadIdx.x < 32) {
    tdm_load_tile(shbase + (uint32_t)(lrow0 * NYc) * 4u,
                  (uint64_t)(uintptr_t)(u1b + (size_t)gstart * NYc),
                  (uint32_t)nrows);
    __builtin_amdgcn_s_wait_tensorcnt(0);
  }
  __syncthreads();      // publish LDS tile (and halo zeros) to all waves

  const float A2  = *A2p;
  const float ih2 = *invh2p;
  const int   sx  = *sxp;
  const int   sy  = *syp;

  for (int r = 0; r < TILE; ++r) {
    const int x   = x0 + r;
    const int row = (r + 1) * NYc;
    const int idx = x * NYc + c;

    const float uc = sh[row + c];
    const float up = sh[row - NYc + c];
    const float dn = sh[row + NYc + c];
    const float lf = (c > 0)       ? sh[row + c - 1] : 0.0f;
    const float rt = (c < NYc - 1) ? sh[row + c + 1] : 0.0f;

    const float lap = ih2 * (up + dn + lf + rt - 4.0f * uc);
    float v = A1[idx] * (A2 * uc - A3[idx] * un2b[idx] + c2[idx] * lap);
    if (x == sx && c == sy) v += xsrc[b * T + t];   // source injection

    unwb[idx]         = v;   // ping-pong state (L2-resident, regular store)
    sres[r * NYc + c] = v;   // stage for NT output stream
  }

  // Output tile is one contiguous 16 KB region: rows x0..x0+15.
  // Stream it with 4 x b128 async LDS->global NT stores per lane.
  __syncthreads();   // cross-lane LDS visibility for the wide chunks
  const uint64_t obase =
      (uint64_t)(uintptr_t)(out + ((size_t)b * T + t) * plane + (size_t)x0 * NYc);
  const uint32_t rbase = (uint32_t)(size_t)(&sres[0]);
  for (int k = 0; k < 4; ++k) {
    const uint32_t byte_off = (uint32_t)(k * NYc + c) * 16u;
    async_store_nt_b128(obase + byte_off, rbase + byte_off);
  }
  wait_asynccnt0();     // stores complete before the wave retires
}

extern "C" void kernel_launch(void* const* d_in, const int* in_sizes, int n_in,
                              void* d_out, int out_size, void* d_ws, size_t ws_size,
                              hipStream_t stream) {
  const float* x   = (const float*)d_in[0];  // (B,T)
  const float* c2  = (const float*)d_in[1];  // (256,256)
  const float* A1  = (const float*)d_in[2];  // (256,256)
  const float* A3  = (const float*)d_in[3];  // (256,256)
  const float* A2  = (const float*)d_in[4];  // scalar
  const float* ih2 = (const float*)d_in[5];  // scalar
  const int*   sx  = (const int*)d_in[6];
  const int*   sy  = (const int*)d_in[7];

  const int    T     = in_sizes[0] / BATCH;            // 300
  const size_t plane = (size_t)NXc * NYc;              // 65536

  // Ping-pong wave state in workspace: 2 * 4 * 256 * 256 * 4B = 2 MB.
  float* P0 = (float*)d_ws;
  float* P1 = P0 + (size_t)BATCH * plane;
  hipMemsetAsync(d_ws, 0, 2 * (size_t)BATCH * plane * sizeof(float), stream);

  dim3 grid(NXc / TILE, BATCH);   // 16 row-blocks x 4 batches
  dim3 block(NYc);                // 256 threads = 8 waves (wave32)

  for (int t = 0; t < T; ++t) {
    const float* un1 = (t & 1) ? P1 : P0;  // result of previous step
    float*       un2 = (t & 1) ? P0 : P1;  // two steps back; overwritten with un
    WaveCell_step_kernel<<<grid, block, 0, stream>>>(
        un1, un2, un2, (float*)d_out,
        A1, A3, c2, x, A2, ih2, sx, sy, t, T);
  }
}